// TreeLSTMLayer_46462956208328
// MI455X (gfx1250) — compile-verified
//
#include <hip/hip_runtime.h>

// Child-Sum Tree-LSTM, complete 4-ary tree, 9 levels, H = E = 512.
// f32 WMMA (V_WMMA_F32_16X16X4_F32); LDS-staged A tiles shared by 4 waves;
// i/o/u gates fused into one triple-accumulator GEMM; f-gate via the same
// template with the row>>2 A1 mapping (children contiguous in prev level).

#define HD 512
#define N_NODES 87381
#define ROWS_MAX 16384          // largest M any GEMM sees (level 7 n, level 6 4n)
#define LDS_PITCH 68            // 64 + 4 pad: stride mod 64 banks = 4 -> no conflicts

typedef __attribute__((ext_vector_type(2))) float v2f;
typedef __attribute__((ext_vector_type(8))) float v8f;

__device__ __forceinline__ float sigmoidf_(float x) {
    return 1.0f / (1.0f + __expf(-x));
}

struct GateArgs {
    const float* W;   // [512 x 512] row-major, applied to A1
    const float* U;   // [512 x 512] row-major, applied to A2
    const float* b;   // [512]
    float*       D;   // [M x 512] output (post-activation)
    int          act; // 0 = sigmoid, 1 = tanh
};

// ---------------------------------------------------------------- zero fill
__global__ void __launch_bounds__(256)
zero_f32(float4* __restrict__ p, long n4) {
    long i = (long)blockIdx.x * blockDim.x + threadIdx.x;
    long stride = (long)gridDim.x * blockDim.x;
    float4 z; z.x = z.y = z.z = z.w = 0.0f;
    for (; i < n4; i += stride) p[i] = z;
}

// ------------------------------------------------- embedding gather + child-sum
// X[k, :]  = emb[tok[tok_off+k], :]
// Ht[k, :] = sum_{b<4} hchild[4k+b, :]        (hchild = prev level h in d_out)
__global__ void __launch_bounds__(256)
gather_childsum(const float* __restrict__ emb, const int* __restrict__ tok,
                int tok_off, const float* __restrict__ hchild,
                float4* __restrict__ X, float4* __restrict__ Ht, int n) {
    int idx = blockIdx.x * blockDim.x + threadIdx.x;     // over n * 128 float4s
    if (idx >= n * (HD / 4)) return;
    int k  = idx >> 7;
    int j4 = (idx & 127) << 2;
    X[idx] = *(const float4*)(emb + (long)tok[tok_off + k] * HD + j4);
    const float* hb = hchild + ((long)(k << 2)) * HD + j4;
    float4 a = *(const float4*)(hb);
    float4 b = *(const float4*)(hb + HD);
    float4 c = *(const float4*)(hb + 2 * HD);
    float4 d = *(const float4*)(hb + 3 * HD);
    float4 s;
    s.x = a.x + b.x + c.x + d.x;
    s.y = a.y + b.y + c.y + d.y;
    s.z = a.z + b.z + c.z + d.z;
    s.w = a.w + b.w + c.w + d.w;
    Ht[idx] = s;
}

// -------------------------------------------------------- multi-gate WMMA GEMM
// For g < NG:  D_g[r, c] = act_g( sum_k A1[r >> a1shift, k] * W_g[k, c]
//                                + sum_k A2[r,           k] * U_g[k, c] + b_g[c] )
// N = K1 = K2 = 512. Block: 128 threads (4 waves), 64x64 output tile.
// A chunks (64 rows x 64 k) staged in LDS, shared by all 4 waves; each staged
// chunk feeds 4*NG WMMAs per k-step. Wave w owns columns [blkN+16w, +16).
template <int NG>
__global__ void __launch_bounds__(128)
gemm_multi(const float* __restrict__ A1, int a1shift,
           const float* __restrict__ A2,
           GateArgs g0, GateArgs g1, GateArgs g2, int M) {
    __shared__ __align__(16) float As[64 * LDS_PITCH];

    const int tid  = threadIdx.x;
    const int lane = tid & 31;
    const int wave = tid >> 5;
    const int m15  = lane & 15;
    const int kb   = (lane >> 4) << 1;          // 0 (lanes 0-15) / 2 (lanes 16-31)
    const int blockM = blockIdx.y * 64;
    const int ncol   = blockIdx.x * 64 + wave * 16 + m15;

    const GateArgs gs[3] = {g0, g1, g2};

    v8f acc[NG][4];
    const v8f vzero = {0.f, 0.f, 0.f, 0.f, 0.f, 0.f, 0.f, 0.f};
#pragma unroll
    for (int g = 0; g < NG; ++g)
#pragma unroll
        for (int t = 0; t < 4; ++t) acc[g][t] = vzero;

    // pass 0: A1 (row >> a1shift) against W; pass 1: A2 against U
    for (int pass = 0; pass < 2; ++pass) {
        const float* A = pass ? A2 : A1;
        const int shift = pass ? 0 : a1shift;
        for (int kc = 0; kc < HD; kc += 64) {
            __syncthreads();
            // stage 64 rows x 64 k-floats: 1024 float4 loads over 128 threads
#pragma unroll
            for (int i = 0; i < 8; ++i) {
                int t2  = i * 128 + tid;
                int row = t2 >> 4;
                int c4  = (t2 & 15) << 2;
                int gr  = blockM + row;
                if (gr > M - 1) gr = M - 1;     // clamp; dup rows dropped at store
                gr >>= shift;
                *(float4*)&As[row * LDS_PITCH + c4] =
                    *(const float4*)(A + (long)gr * HD + kc + c4);
            }
            __syncthreads();
            for (int kl = 0; kl < 64; kl += 4) {
                v2f a[4];
#pragma unroll
                for (int t = 0; t < 4; ++t)
                    a[t] = *(const v2f*)&As[(t * 16 + m15) * LDS_PITCH + kl + kb];
                const int krow = kc + kl + kb;
#pragma unroll
                for (int g = 0; g < NG; ++g) {
                    const float* Wm = pass ? gs[g].U : gs[g].W;
                    v2f b;
                    b.x = Wm[(long)krow * HD + ncol];
                    b.y = Wm[(long)(krow + 1) * HD + ncol];
#pragma unroll
                    for (int t = 0; t < 4; ++t)
                        acc[g][t] = __builtin_amdgcn_wmma_f32_16x16x4_f32(
                            false, a[t], false, b, (short)0, acc[g][t],
                            false, false);
                }
            }
        }
    }

    // epilogue: bias + activation + bounds-checked store
    const int rhi = (lane >> 4) << 3;           // +8 rows for lanes 16-31
#pragma unroll
    for (int g = 0; g < NG; ++g) {
        const float bv = gs[g].b[ncol];
#pragma unroll
        for (int t = 0; t < 4; ++t) {
#pragma unroll
            for (int v = 0; v < 8; ++v) {
                int row = blockM + t * 16 + rhi + v;
                if (row < M) {
                    float x = acc[g][t][v] + bv;
                    gs[g].D[(long)row * HD + ncol] =
                        gs[g].act ? tanhf(x) : sigmoidf_(x);
                }
            }
        }
    }
}

// ------------------------------------------------------------- combine
// c[k] = i*u + sum_b f[4k+b] * c_prev[4k+b];  h[k] = o * tanh(c[k])
__global__ void __launch_bounds__(256)
combine(const float4* __restrict__ Gi, const float4* __restrict__ Go,
        const float4* __restrict__ Gu, const float* __restrict__ F,
        const float* __restrict__ cprev, float4* __restrict__ ccur,
        float4* __restrict__ hout, int n, int has_children) {
    int idx = blockIdx.x * blockDim.x + threadIdx.x;     // over n * 128 float4s
    if (idx >= n * (HD / 4)) return;
    int k  = idx >> 7;
    int j4 = (idx & 127) << 2;
    float4 gi = Gi[idx], gu = Gu[idx], go = Go[idx];
    float4 cc;
    cc.x = gi.x * gu.x; cc.y = gi.y * gu.y;
    cc.z = gi.z * gu.z; cc.w = gi.w * gu.w;
    if (has_children) {
        long base = ((long)(k << 2)) * HD + j4;
#pragma unroll
        for (int b = 0; b < 4; ++b) {
            float4 f = *(const float4*)(F + base + b * HD);
            float4 cp = *(const float4*)(cprev + base + b * HD);
            cc.x += f.x * cp.x; cc.y += f.y * cp.y;
            cc.z += f.z * cp.z; cc.w += f.w * cp.w;
        }
    }
    ccur[idx] = cc;
    float4 h;
    h.x = go.x * tanhf(cc.x); h.y = go.y * tanhf(cc.y);
    h.z = go.z * tanhf(cc.z); h.w = go.w * tanhf(cc.w);
    hout[idx] = h;
}

// ------------------------------------------------------------- root h,c tail
__global__ void __launch_bounds__(512)
root_copy(float* __restrict__ out, const float* __restrict__ croot) {
    int j = threadIdx.x;
    const long tail = (long)N_NODES * HD;
    out[tail + j]      = out[(long)(N_NODES - 1) * HD + j];  // root h
    out[tail + HD + j] = croot[j];                           // root c
}

extern "C" void kernel_launch(void* const* d_in, const int* in_sizes, int n_in,
                              void* d_out, int out_size, void* d_ws, size_t ws_size,
                              hipStream_t stream) {
    const int*   tok = (const int*)d_in[0];
    const float* emb = (const float*)d_in[1];
    const float* Wi = (const float*)d_in[2];
    const float* Ui = (const float*)d_in[3];
    const float* bi = (const float*)d_in[4];
    const float* Wf = (const float*)d_in[5];
    const float* Uf = (const float*)d_in[6];
    const float* bf = (const float*)d_in[7];
    const float* Wo = (const float*)d_in[8];
    const float* Uo = (const float*)d_in[9];
    const float* bo = (const float*)d_in[10];
    const float* Wu = (const float*)d_in[11];
    const float* Uu = (const float*)d_in[12];
    const float* bu = (const float*)d_in[13];
    float* out = (float*)d_out;

    // workspace carve-up: 8 regions of ROWS_MAX x 512 f32 (32 MB each, 256 MB)
    const size_t R = (size_t)ROWS_MAX * HD;
    float* ws = (float*)d_ws;
    float* X  = ws + 0 * R;
    float* Ht = ws + 1 * R;
    float* Gi = ws + 2 * R;
    float* Go = ws + 3 * R;
    float* Gu = ws + 4 * R;
    float* F  = ws + 5 * R;
    float* cA = ws + 6 * R;
    float* cB = ws + 7 * R;

    // level tables (level 0 = root ... level 8 = leaves)
    const int  lvl_n[9]   = {1, 4, 16, 64, 256, 1024, 4096, 16384, 65536};
    const int  tok_off[9] = {0, 1, 5, 21, 85, 341, 1365, 5461, 21845};
    // row offset of level l within hidden_states (leaves first, then l=7..0)
    const long hs_off[9]  = {87380, 87376, 87360, 87296, 87040, 86016,
                             81920, 65536, 0};

    // leaves: h = 0 in the output (reference includes 65536 zero rows)
    zero_f32<<<4096, 256, 0, stream>>>((float4*)out, (long)65536 * (HD / 4));

    float* cprev = cA;
    float* ccur  = cB;

    for (int l = 7; l >= 0; --l) {
        const int  n        = lvl_n[l];
        const long childoff = hs_off[l + 1] * HD;   // children h rows in d_out
        const int  hasch    = (l < 7);              // level 7 children are leaves

        int eg = (n * (HD / 4) + 255) / 256;
        gather_childsum<<<eg, 256, 0, stream>>>(emb, tok, tok_off[l],
                                                out + childoff,
                                                (float4*)X, (float4*)Ht, n);

        GateArgs ai = {Wi, Ui, bi, Gi, 0};
        GateArgs ao = {Wo, Uo, bo, Go, 0};
        GateArgs au = {Wu, Uu, bu, Gu, 1};

        dim3 g(HD / 64, (n + 63) / 64);
        gemm_multi<3><<<g, 128, 0, stream>>>(X, 0, Ht, ai, ao, au, n);

        if (hasch) {
            // f gate over all 4n children: row r -> X[r>>2] @ Wf + h_prev[r] @ Uf
            GateArgs af = {Wf, Uf, bf, F, 0};
            dim3 gf(HD / 64, (4 * n + 63) / 64);
            gemm_multi<1><<<gf, 128, 0, stream>>>(X, 2, out + childoff,
                                                  af, af, af, 4 * n);
        }

        combine<<<eg, 256, 0, stream>>>((const float4*)Gi, (const float4*)Go,
                                        (const float4*)Gu, F, cprev,
                                        (float4*)ccur,
                                        (float4*)(out + hs_off[l] * HD),
                                        n, hasch);

        float* t = cprev; cprev = ccur; ccur = t;   // ccur -> becomes c_prev
    }

    // after the loop, cprev holds the root c (written at l = 0)
    root_copy<<<1, 512, 0, stream>>>(out, cprev);
}